// CostMapLayer_88734024335487
// MI455X (gfx1250) — compile-verified
//
#include <hip/hip_runtime.h>

typedef float v4f __attribute__((ext_vector_type(4)));

#define TPB 256
#define PPT 4   // points per thread

// Order-preserving float <-> uint mapping so global_atomic_min_u32 implements float min.
__device__ __forceinline__ unsigned enc_ord(float f) {
    unsigned u = __float_as_uint(f);
    return (u & 0x80000000u) ? ~u : (u | 0x80000000u);
}
__device__ __forceinline__ float dec_ord(unsigned u) {
    return (u & 0x80000000u) ? __uint_as_float(u & 0x7FFFFFFFu) : __uint_as_float(~u);
}

__global__ void costmap_init(unsigned* __restrict__ cnt,
                             unsigned* __restrict__ om, int n) {
    int i = blockIdx.x * blockDim.x + threadIdx.x;
    if (i < n) {
        cnt[i] = 0u;
        om[i]  = 0xFFFFFFFFu;   // > enc_ord(any real float); only read when cnt>0
    }
}

__device__ __forceinline__ void scatter_point(float x, float y, float c, int b,
                                              int H, int W, int HW,
                                              unsigned* __restrict__ cnt,
                                              unsigned* __restrict__ om) {
    int ix = (int)floorf(x + 0.5f);
    int iy = (int)floorf(y + 0.5f);
    if (ix < 0 || ix >= W || iy < 0 || iy >= H) return;   // invalid = no-op in reference
    int flat = b * HW + iy * W + ix;
    atomicAdd(&cnt[flat], 1u);                 // global_atomic_add_u32 (L2-resident map)
    atomicMin(&om[flat], enc_ord(c));          // global_atomic_min_u32 (ordered-float min)
}

__global__ void costmap_scatter(const float* __restrict__ coords,
                                const float* __restrict__ costs,
                                const int* __restrict__ Hp,
                                const int* __restrict__ Wp,
                                unsigned* __restrict__ cnt,
                                unsigned* __restrict__ om,
                                int nPoints, int mapElems) {
    const int H  = *Hp;
    const int W  = *Wp;
    const int HW = H * W;
    const int B  = mapElems / HW;          // uniform scalar math
    const int N  = nPoints / B;

    const int p0 = (blockIdx.x * TPB + threadIdx.x) * PPT;
    if (p0 >= nPoints) return;

    // Batch index: one *uniform* division per block (scalar path), then a cheap
    // per-lane correction; avoids a ~30-instruction u32 divide in every lane.
    const int blockStart = blockIdx.x * (TPB * PPT);
    const int bBase      = blockStart / N;          // uniform -> SALU
    int b   = bBase;
    int rem = p0 - bBase * N;                       // rem < N + TPB*PPT
    while (rem >= N) { rem -= N; ++b; }             // almost always 0 iterations

    if (p0 + (PPT - 1) < nPoints) {
        // Streamed once: nontemporal so the 96MB point stream doesn't evict the
        // 16MB L2-resident map. Two B128 coord loads + one B128 cost load.
        v4f ca = __builtin_nontemporal_load((const v4f*)(coords + (size_t)p0 * 2));
        v4f cb = __builtin_nontemporal_load((const v4f*)(coords + (size_t)p0 * 2 + 4));
        v4f k  = __builtin_nontemporal_load((const v4f*)(costs + (size_t)p0));
        float xs[PPT] = {ca.x, ca.z, cb.x, cb.z};
        float ys[PPT] = {ca.y, ca.w, cb.y, cb.w};
        float cs[PPT] = {k.x, k.y, k.z, k.w};
#pragma unroll
        for (int j = 0; j < PPT; ++j) {
            scatter_point(xs[j], ys[j], cs[j], b, H, W, HW, cnt, om);
            if (++rem == N) { rem = 0; ++b; }       // cross batch boundary
        }
    } else {
        for (int p = p0; p < nPoints; ++p) {        // tail (< PPT points)
            float x = coords[(size_t)p * 2];
            float y = coords[(size_t)p * 2 + 1];
            float c = costs[p];
            scatter_point(x, y, c, b, H, W, HW, cnt, om);
            if (++rem == N) { rem = 0; ++b; }
        }
    }
}

__global__ void costmap_finalize(const unsigned* __restrict__ cnt,
                                 const unsigned* __restrict__ om,
                                 const float* __restrict__ defc,
                                 float* __restrict__ out, int mapElems) {
    int i = blockIdx.x * blockDim.x + threadIdx.x;
    if (i >= mapElems) return;
    unsigned c = cnt[i];
    float cost = (c > 0u) ? dec_ord(om[i]) : defc[0];
    __builtin_nontemporal_store(cost, out + i);                           // cost map
    __builtin_nontemporal_store((float)((int)c - 1), out + mapElems + i); // mask = count-1
}

extern "C" void kernel_launch(void* const* d_in, const int* in_sizes, int n_in,
                              void* d_out, int out_size, void* d_ws, size_t ws_size,
                              hipStream_t stream) {
    const float* coords = (const float*)d_in[0];   // [B,N,2] f32
    const float* costs  = (const float*)d_in[1];   // [B,N]   f32
    const float* defc   = (const float*)d_in[2];   // [1]     f32
    const int*   Hp     = (const int*)d_in[3];     // device scalar
    const int*   Wp     = (const int*)d_in[4];     // device scalar

    const int nPoints  = in_sizes[1];       // B*N
    const int mapElems = out_size / 2;      // B*H*W (outputs: cost ++ mask)

    unsigned* cnt = (unsigned*)d_ws;        // [mapElems] u32
    unsigned* om  = cnt + mapElems;         // [mapElems] u32 (ordered-min keys)
    float*    out = (float*)d_out;

    const int gMap = (mapElems + TPB - 1) / TPB;
    costmap_init<<<gMap, TPB, 0, stream>>>(cnt, om, mapElems);

    const int nChunks = (nPoints + PPT - 1) / PPT;
    const int gScat   = (nChunks + TPB - 1) / TPB;
    costmap_scatter<<<gScat, TPB, 0, stream>>>(coords, costs, Hp, Wp,
                                               cnt, om, nPoints, mapElems);

    costmap_finalize<<<gMap, TPB, 0, stream>>>(cnt, om, defc, out, mapElems);
}